// NormConnectionLaplacianBuilder_18459769438525
// MI455X (gfx1250) — compile-verified
//
#include <hip/hip_runtime.h>
#include <hip/hip_bf16.h>

typedef __attribute__((ext_vector_type(2))) float v2f;
typedef __attribute__((ext_vector_type(8))) float v8f;

#define D4 4
#define WPB 8   // waves per block in the WMMA kernel

// ---------------------------------------------------------------------------
// Kernel 1: exclusive prefix sum of integer degrees (single block, 1024 thr).
// cdeg[p] = sum_{p'<p} deg[p'],  cdeg[N] = 2E.
// Row start for global row (p,i) is then 16*cdeg[p] + 4*p + i*(4*deg[p]+1).
// ---------------------------------------------------------------------------
__global__ __launch_bounds__(1024) void scan_kernel(const float* __restrict__ deg,
                                                    int* __restrict__ cdeg, int N) {
    __shared__ int sh[1024];
    __shared__ int carry;
    if (threadIdx.x == 0) carry = 0;
    __syncthreads();
    for (int base = 0; base < N; base += 1024) {
        int idx = base + (int)threadIdx.x;
        int d = (idx < N) ? (int)(deg[idx] + 0.5f) : 0;
        sh[threadIdx.x] = d;
        __syncthreads();
        for (int off = 1; off < 1024; off <<= 1) {  // Hillis-Steele inclusive scan
            int v = (threadIdx.x >= (unsigned)off) ? sh[threadIdx.x - off] : 0;
            __syncthreads();
            sh[threadIdx.x] += v;
            __syncthreads();
        }
        int incl = sh[threadIdx.x];
        int c = carry;
        __syncthreads();
        if (idx < N) cdeg[idx] = c + incl - d;      // exclusive
        if (threadIdx.x == 1023) carry = c + incl;
        __syncthreads();
    }
    if (threadIdx.x == 0) cdeg[N] = carry;
}

// ---------------------------------------------------------------------------
// Kernel 2: fill per-node neighbor lists (CSR) from both edge directions.
// Slot order is nondeterministic (atomic) but only value-ranks are consumed,
// so the final output is deterministic (neighbors are distinct).
// ---------------------------------------------------------------------------
__global__ void fill_kernel(const int* __restrict__ a, const int* __restrict__ b,
                            const int* __restrict__ cdeg, int* __restrict__ fill,
                            int* __restrict__ nbrQ, int E) {
    int t = blockIdx.x * blockDim.x + threadIdx.x;
    if (t >= 2 * E) return;
    int p, q;
    if (t < E) { p = a[t]; q = b[t]; } else { p = b[t - E]; q = a[t - E]; }
    int slot = cdeg[p] + atomicAdd(&fill[p], 1);
    nbrQ[slot] = q;
}

// ---------------------------------------------------------------------------
// Kernel 3: rank of q among sorted neighbors of p, for each directed record.
// rankOf[e]   = rank of b[e] among neighbors of a[e]
// rankOf[e+E] = rank of a[e] among neighbors of b[e]
// ---------------------------------------------------------------------------
__global__ void rank_kernel(const int* __restrict__ a, const int* __restrict__ b,
                            const int* __restrict__ cdeg, const int* __restrict__ nbrQ,
                            int* __restrict__ rankOf, int E) {
    int t = blockIdx.x * blockDim.x + threadIdx.x;
    if (t >= 2 * E) return;
    int p, q;
    if (t < E) { p = a[t]; q = b[t]; } else { p = b[t - E]; q = a[t - E]; }
    int s = cdeg[p];
    int d = cdeg[p + 1] - s;
    int c = 0;
    for (int k = 0; k < d; k++) c += (nbrQ[s + k] < q) ? 1 : 0;
    rankOf[t] = c;
}

// ---------------------------------------------------------------------------
// Kernel 4: Cayley transform (one 4x4 orthogonal map per lane -> 32 maps =
// 16 edges per wave), then 4x V_WMMA_F32_16X16X4_F32 per wave computing the
// -(L^T R) transport blocks for 4 edges each (diagonal 4x4 blocks of the
// 16x16 f32 accumulator), written straight into the saved_tril_maps output.
// ---------------------------------------------------------------------------
__global__ __launch_bounds__(32 * WPB) void cayley_wmma_kernel(
    const float* __restrict__ params, const int* __restrict__ left_idx,
    const int* __restrict__ right_idx, float* __restrict__ saved, int E) {
    __shared__ float S[WPB][32][16];
    const int lane = threadIdx.x & 31;
    const int wv   = threadIdx.x >> 5;
    const long gw  = (long)blockIdx.x * WPB + wv;
    const int eBase = (int)(gw * 16);

    // --- Phase 1: Cayley. Lane t<16 -> left map of edge eBase+t,
    //              lane t>=16 -> right map of edge eBase+t-16.
    int eLoc = lane & 15;
    int eIdx = eBase + eLoc;
    bool valid = (eIdx < E);
    int dIdx = 0;
    if (valid) dIdx = (lane >= 16) ? right_idx[eIdx] : left_idx[eIdx];
    const float* pp = params + (long)dIdx * 10;
    float P[10];
#pragma unroll
    for (int k = 0; k < 10; k++) P[k] = valid ? pp[k] : 0.0f;
    // H = A/2, A = X - X^T, X lower-tri from params (row-major tril order).
    // Diagonal params cancel; only 6 off-diagonals survive.
    float h1 = 0.5f * P[1], h3 = 0.5f * P[3], h4 = 0.5f * P[4];
    float h6 = 0.5f * P[6], h7 = 0.5f * P[7], h8 = 0.5f * P[8];
    float H[4][4] = {{0.f, -h1, -h3, -h6},
                     {h1, 0.f, -h4, -h7},
                     {h3, h4, 0.f, -h8},
                     {h6, h7, h8, 0.f}};
    // Solve (I-H) M = (I+H) by Gauss-Jordan (I-H has SPD symmetric part).
    float G[4][8];
#pragma unroll
    for (int i = 0; i < 4; i++)
#pragma unroll
        for (int j = 0; j < 4; j++) {
            float id = (i == j) ? 1.0f : 0.0f;
            G[i][j]     = id - H[i][j];
            G[i][4 + j] = id + H[i][j];
        }
#pragma unroll
    for (int k = 0; k < 4; k++) {
        float inv = 1.0f / G[k][k];
#pragma unroll
        for (int j = 0; j < 8; j++) G[k][j] *= inv;
#pragma unroll
        for (int r = 0; r < 4; r++) {
            if (r == k) continue;
            float f = G[r][k];
#pragma unroll
            for (int j = 0; j < 8; j++) G[r][j] -= f * G[k][j];
        }
    }
#pragma unroll
    for (int i = 0; i < 4; i++)
#pragma unroll
        for (int j = 0; j < 4; j++) S[wv][lane][i * 4 + j] = G[i][4 + j];
    __syncthreads();

    // --- Phase 2: 4 WMMAs, each covering 4 edges.
    // A (16x4 f32): row m = 4*eg+i holds L_e[k][i] (k=0..3).
    //   Layout: lane l -> M = l%16, VGPR v -> K = 2*(l/16)+v.
    // B (4x16 f32): col n = 4*eg+j holds R_e[k][j].
    // D (16x16 f32): diagonal 4x4 blocks = L^T R per edge.
    const float* Sw = &S[wv][0][0];
    const int n  = lane & 15;
    const int kh = lane >> 4;     // 0: K=0,1  1: K=2,3
    const int sub = n >> 2;       // edge-within-group 0..3
    const int ii  = n & 3;        // i for A, j for B (same decomposition)
#pragma unroll
    for (int g = 0; g < 4; g++) {
        int eL = 4 * g + sub;
        v2f aop, bop;
        aop.x = Sw[eL * 16 + (2 * kh + 0) * 4 + ii];        // L[k][i]
        aop.y = Sw[eL * 16 + (2 * kh + 1) * 4 + ii];
        bop.x = Sw[(16 + eL) * 16 + (2 * kh + 0) * 4 + ii]; // R[k][j]
        bop.y = Sw[(16 + eL) * 16 + (2 * kh + 1) * 4 + ii];
        v8f c = {0.f, 0.f, 0.f, 0.f, 0.f, 0.f, 0.f, 0.f};
        c = __builtin_amdgcn_wmma_f32_16x16x4_f32(false, aop, false, bop,
                                                  (short)0, c, false, false);
        // Extract diagonal block (eg,eg): rows m=4*eg..4*eg+3, col n=4*eg+j.
        // Lanes 0-15 hold rows 0-7 in vgprs 0-7; lanes 16-31 hold rows 8-15.
        bool act = ((lane < 16) == (n < 8));
        if (act) {
            int vb = (sub & 1) * 4;
            float r0 = vb ? c[4] : c[0];
            float r1 = vb ? c[5] : c[1];
            float r2 = vb ? c[6] : c[2];
            float r3 = vb ? c[7] : c[3];
            int e = eBase + eL;
            if (e < E) {
                float* dst = saved + (long)e * 16 + ii;  // column j, rows i=0..3
                dst[0]  = -r0;
                dst[4]  = -r1;
                dst[8]  = -r2;
                dst[12] = -r3;
            }
        }
    }
}

// ---------------------------------------------------------------------------
// Kernel 5: scatter the 16 normalized values per edge to their analytic
// sorted positions (tril block above diag col-wise, mirrored triu block).
// ---------------------------------------------------------------------------
__global__ void scatter_kernel(const int* __restrict__ a, const int* __restrict__ b,
                               const int* __restrict__ cdeg, const int* __restrict__ rankOf,
                               const float* __restrict__ saved,
                               float* __restrict__ rows, float* __restrict__ cols,
                               float* __restrict__ w, int E) {
    int e = blockIdx.x * blockDim.x + threadIdx.x;
    if (e >= E) return;
    int av = a[e], bv = b[e];
    int sa = cdeg[av], da = cdeg[av + 1] - sa;
    int sb = cdeg[bv], db = cdeg[bv + 1] - sb;
    float scale = (1.0f / sqrtf((float)da + 1.0f)) * (1.0f / sqrtf((float)db + 1.0f));
    int r_ab = rankOf[e];
    int r_ba = rankOf[e + E];
    int base1 = 16 * sa + 4 * av;  // nodeStart(a)
    int base2 = 16 * sb + 4 * bv;  // nodeStart(b)
    int la = 4 * da + 1, lb = 4 * db + 1;
    const float* sv = saved + (long)e * 16;
#pragma unroll
    for (int i = 0; i < 4; i++) {
#pragma unroll
        for (int j = 0; j < 4; j++) {
            float nd = scale * sv[i * 4 + j];
            // entry (a*4+i, b*4+j): block after diag in row of node a (b > a)
            int pos1 = base1 + i * la + 4 * r_ab + 1 + j;
            rows[pos1] = (float)(4 * av + i);
            cols[pos1] = (float)(4 * bv + j);
            w[pos1] = nd;
            // mirrored entry (b*4+j, a*4+i): block before diag in row of b
            int pos2 = base2 + j * lb + 4 * r_ba + i;
            rows[pos2] = (float)(4 * bv + j);
            cols[pos2] = (float)(4 * av + i);
            w[pos2] = nd;
        }
    }
}

// ---------------------------------------------------------------------------
// Kernel 6: diagonal entries deg/(deg+1), placed at rank 4*lowrank(p).
// ---------------------------------------------------------------------------
__global__ void diag_kernel(const int* __restrict__ cdeg, const int* __restrict__ nbrQ,
                            float* __restrict__ rows, float* __restrict__ cols,
                            float* __restrict__ w, int N) {
    int p = blockIdx.x * blockDim.x + threadIdx.x;
    if (p >= N) return;
    int s = cdeg[p];
    int d = cdeg[p + 1] - s;
    int c = 0;
    for (int k = 0; k < d; k++) c += (nbrQ[s + k] < p) ? 1 : 0;
    float degf = (float)d;
    float val = degf / (degf + 1.0f);
    int base = 16 * s + 4 * p;
    int l = 4 * d + 1;
#pragma unroll
    for (int i = 0; i < 4; i++) {
        int pos = base + i * l + 4 * c;
        float rc = (float)(4 * p + i);
        rows[pos] = rc;
        cols[pos] = rc;
        w[pos] = val;
    }
}

extern "C" void kernel_launch(void* const* d_in, const int* in_sizes, int n_in,
                              void* d_out, int out_size, void* d_ws, size_t ws_size,
                              hipStream_t stream) {
    const float* map_params = (const float*)d_in[0];
    const float* deg        = (const float*)d_in[1];
    const int*   left_idx   = (const int*)d_in[2];
    const int*   right_idx  = (const int*)d_in[3];
    const int*   tril_row   = (const int*)d_in[4];  // a (src < dst)
    const int*   tril_col   = (const int*)d_in[5];  // b
    // d_in[6] tril_indices / d_in[7] diag_indices: positions derived analytically

    const int N = in_sizes[1];
    const int E = in_sizes[2];
    const long M = 2L * E * 16 + 4L * N;  // total sparse entries

    float* out   = (float*)d_out;
    float* rows  = out;          // edge_index row 0 (sorted)
    float* cols  = out + M;      // edge_index row 1
    float* w     = out + 2 * M;  // weights
    float* saved = out + 3 * M;  // saved_tril_maps [E,4,4]

    int* cdeg   = (int*)d_ws;          // N+1
    int* fill   = cdeg + (N + 1);      // N
    int* nbrQ   = fill + N;            // 2E
    int* rankOf = nbrQ + 2L * E;       // 2E

    scan_kernel<<<1, 1024, 0, stream>>>(deg, cdeg, N);
    hipMemsetAsync(fill, 0, (size_t)N * sizeof(int), stream);
    int t2 = 2 * E;
    fill_kernel<<<(t2 + 255) / 256, 256, 0, stream>>>(tril_row, tril_col, cdeg, fill, nbrQ, E);
    rank_kernel<<<(t2 + 255) / 256, 256, 0, stream>>>(tril_row, tril_col, cdeg, nbrQ, rankOf, E);

    int nWaves  = (E + 15) / 16;
    int nBlocks = (nWaves + WPB - 1) / WPB;
    cayley_wmma_kernel<<<nBlocks, 32 * WPB, 0, stream>>>(map_params, left_idx, right_idx, saved, E);

    scatter_kernel<<<(E + 255) / 256, 256, 0, stream>>>(tril_row, tril_col, cdeg, rankOf,
                                                        saved, rows, cols, w, E);
    diag_kernel<<<(N + 255) / 256, 256, 0, stream>>>(cdeg, nbrQ, rows, cols, w, N);
}